// AdaptiveQuantization_31490700214749
// MI455X (gfx1250) — compile-verified
//
#include <hip/hip_runtime.h>

// Adaptive quantization (training mode): per-element bin search over 61
// adaptive bins + uniform sample inside the bin.
//
// MI455X reasoning: 12 B/element of mandatory HBM traffic vs ~30 FLOP/element
// => bandwidth-bound (floor ~1.6us at 23.3 TB/s, 37.7 MB total). WMMA on the
// reference's (N,61) one-hot matmul would multiply FLOPs by 61x and traffic
// by ~40x, so the optimal mix is b128 NT streaming + LDS table lookups.
// This revision moves the first 3 (wave-uniform) binary-search levels from
// the LDS pipe to the VALU pipe (v_cndmask selects) to cut DS ops ~2x, and
// processes 8 elements/thread to amortize the per-thread table preloads.

#define EXTREMA 30      // L
#define NBINS   61      // 2L + 1
#define NBPAD   64      // power-of-two padded search array

typedef float v4f __attribute__((ext_vector_type(4)));

__global__ __launch_bounds__(256) void aq_kernel(
    const float* __restrict__ x,
    const float* __restrict__ noise,
    const float* __restrict__ w,
    float* __restrict__ out,
    long long n)
{
    __shared__ float  s_cum[NBINS];      // 61 boundaries
    __shared__ float  s_dist[NBINS - 1]; // 60 half-widths
    __shared__ float  s_avg[NBPAD];      // 60 midpoints, padded with +INF
    __shared__ float4 s_tab[NBINS];      // (avg_left, avg_right, dpl, dpr)

    const int tid = threadIdx.x;

    // ---- build bin tables from w (tiny: 30 elements, SALU-friendly) -----
    if (tid == 0) {
        float run = 0.0f;
        #pragma unroll
        for (int i = 0; i < EXTREMA; ++i) { run += w[i]; s_cum[31 + i] = run; }
        s_cum[30] = 0.0f;
    }
    __syncthreads();
    if (tid < EXTREMA) {
        // cum[j] = -cw[29-j] for j<30 ; cw[i] lives at s_cum[31+i]
        s_cum[tid] = -s_cum[60 - tid];
    }
    __syncthreads();
    if (tid < NBINS - 1) {
        float lo = s_cum[tid], hi = s_cum[tid + 1];
        s_avg[tid]  = 0.5f * (hi + lo);
        s_dist[tid] = 0.5f * (hi - lo);
    }
    if (tid >= NBINS - 1 && tid < NBPAD) s_avg[tid] = __builtin_inff();
    __syncthreads();
    if (tid < NBINS) {
        // faithful to torch: avg_left[0] = -(cum[0] - dist[0]) (sign quirk)
        float al = (tid == 0)         ? -(s_cum[0] - s_dist[0])                 : s_avg[tid - 1];
        float ar = (tid == NBINS - 1) ?  (s_cum[NBINS - 1] + s_dist[NBINS - 2]) : s_avg[tid];
        float dl = (tid == 0)         ? 0.0f : s_dist[tid - 1];
        float dr = (tid == NBINS - 1) ? 0.0f : s_dist[tid];
        s_tab[tid] = make_float4(al, ar, dl, dr);
    }
    __syncthreads();

    // Wave-uniform pivots for the first 3 search levels -> registers.
    const float a31 = s_avg[31];
    const float a15 = s_avg[15], a47 = s_avg[47];
    const float a7  = s_avg[7],  a23 = s_avg[23];
    const float a39 = s_avg[39], a55 = s_avg[55];

    // ---- per-element: 3 ALU + 3 LDS search levels + table gather --------
    auto quant = [&](float xs, float ns) -> float {
        // k = #{ j : avg[j] < xs }, array padded with +INF to 64
        bool c1 = xs > a31;                 // level s=32
        int  k  = c1 ? 32 : 0;
        float p16 = c1 ? a47 : a15;         // level s=16
        bool c2 = xs > p16;
        k += c2 ? 16 : 0;
        float pa = c2 ? a23 : a7;           // level s=8
        float pb = c2 ? a55 : a39;
        float p8 = c1 ? pb : pa;
        k += (xs > p8) ? 8 : 0;
        #pragma unroll
        for (int s = 4; s > 0; s >>= 1) {   // levels s=4,2,1 (bank-conflict-free)
            if (xs > s_avg[k + s - 1]) k += s;
        }
        float4 t  = s_tab[k];               // (al, ar, dpl, dpr)
        int fire  = (xs > t.x) & (xs <= t.y);
        float v0  = fire ? t.z : 0.0f;
        float v1  = fire ? t.w : 0.0f;
        float li  = xs - v0;
        float ri  = xs + v1;
        return fmaf(li - ri, ns, ri);       // (li-ri)*noise + ri
    };

    const long long n4      = n >> 2;
    const long long gid     = (long long)blockIdx.x * blockDim.x + tid;
    const long long gstride = (long long)gridDim.x * blockDim.x;

    const v4f* __restrict__ x4 = (const v4f*)x;
    const v4f* __restrict__ z4 = (const v4f*)noise;
    v4f* __restrict__       o4 = (v4f*)out;

    // Grid is sized to n4/2, so each thread typically handles 2 float4s.
    for (long long i = gid; i < n4; i += gstride) {
        v4f xv = __builtin_nontemporal_load(&x4[i]);   // global_load_b128 th:NT
        v4f zv = __builtin_nontemporal_load(&z4[i]);   // global_load_b128 th:NT
        v4f ov;
        ov.x = quant(xv.x, zv.x);
        ov.y = quant(xv.y, zv.y);
        ov.z = quant(xv.z, zv.z);
        ov.w = quant(xv.w, zv.w);
        __builtin_nontemporal_store(ov, &o4[i]);       // global_store_b128 th:NT
    }
    // scalar tail (N here is divisible by 4, kept for generality)
    for (long long i = (n4 << 2) + gid; i < n; i += gstride) {
        out[i] = quant(x[i], noise[i]);
    }
}

extern "C" void kernel_launch(void* const* d_in, const int* in_sizes, int n_in,
                              void* d_out, int out_size, void* d_ws, size_t ws_size,
                              hipStream_t stream)
{
    const float* x     = (const float*)d_in[0];
    const float* noise = (const float*)d_in[1];
    const float* w     = (const float*)d_in[2];
    float*       out   = (float*)d_out;

    const long long n  = (long long)in_sizes[0];
    const long long n4 = n >> 2;

    // 2 float4s (8 elements) per thread: amortizes table preloads while
    // still launching ~12K wave32s -> saturates 23.3 TB/s HBM.
    long long work = (n4 + 1) >> 1;
    int blocks = (int)((work + 255) / 256);
    if (blocks < 1) blocks = 1;

    aq_kernel<<<blocks, 256, 0, stream>>>(x, noise, w, out, n);
}